// GraphSAGELayer_35802847380153
// MI455X (gfx1250) — compile-verified
//
#include <hip/hip_runtime.h>
#include <hip/hip_bf16.h>

// ---------------------------------------------------------------------------
// GraphSAGE layer for MI455X (gfx1250, wave32).
//   support = segment_sum(x[col], row) / max(deg,1)
//   out     = [x | support] @ W + bias
// Memory/atomic-bound: fp32 end-to-end; GEMM uses V_WMMA_F32_16X16X4_F32.
// ---------------------------------------------------------------------------

typedef __attribute__((ext_vector_type(2))) float v2f;
typedef __attribute__((ext_vector_type(8))) float v8f;

#define D_IN  64
#define D_HID 128
#define D_OUT 64

// ---------------- Kernel 1: zero workspace (support + deg) -----------------
__global__ __launch_bounds__(256) void sage_zero(float* __restrict__ p, int n) {
    int i = blockIdx.x * blockDim.x + threadIdx.x;
    if (i < n) p[i] = 0.0f;
}

// ---------------- Kernel 2: edge scatter-add (one edge per wave32) ---------
__global__ __launch_bounds__(256) void sage_scatter(
    const int* __restrict__ ei,        // [2, E] (row = dst, col = src)
    const float* __restrict__ x,       // [N, 64]
    float* __restrict__ support,       // [N, 64] accumulators
    float* __restrict__ deg,           // [N]
    int n_edges)
{
    int gwave = (int)((blockIdx.x * (unsigned)blockDim.x + threadIdx.x) >> 5);
    int lane  = threadIdx.x & 31;
    if (gwave >= n_edges) return;

    int dst = ei[gwave];
    int src = ei[n_edges + gwave];

    // 32 lanes x 2 floats = one 64-float feature row, coalesced 256B load.
    const float* xs = x + (size_t)src * D_IN;
    float v0 = xs[2 * lane + 0];
    float v1 = xs[2 * lane + 1];

    float* sp = support + (size_t)dst * D_IN;
    atomicAdd(&sp[2 * lane + 0], v0);   // global_atomic_add_f32
    atomicAdd(&sp[2 * lane + 1], v1);
    if (lane == 0) atomicAdd(&deg[dst], 1.0f);
}

// ---------------- Kernel 3: fused mean + concat + GEMM (WMMA f32) ----------
// Block = 128 threads (4 waves). Each block owns a 16-row output tile.
// Wave w computes output columns [16w, 16w+16) with K=128 via 32 WMMAs.
__global__ __launch_bounds__(128) void sage_gemm(
    const float* __restrict__ x,       // [N, 64]
    const float* __restrict__ W,       // [128, 64]
    const float* __restrict__ bias,    // [64]
    const float* __restrict__ support, // [N, 64] (raw sums)
    const float* __restrict__ deg,     // [N]
    float* __restrict__ out,           // [N, 64]
    int n)
{
    // 16 rows x 128 cols of h, padded stride 132 -> conflict-free LDS reads
    // (132 % 64 == 4, so 16 consecutive rows land on distinct banks).
    __shared__ float hs[16][D_HID + 4];

    const int row0 = blockIdx.x * 16;
    const int tid  = threadIdx.x;
    const int lane = tid & 31;
    const int wave = tid >> 5;          // 0..3 -> output column tile

    // ---- Stage 1: build h tile = [x | support/deg] in LDS ----
    {
        int r  = tid >> 3;              // 0..15 : row within tile
        int cg = tid & 7;               // 0..7  : 16-col chunk
        int row = row0 + r;
        bool valid = row < n;
        float invd = 1.0f;
        if (valid) invd = 1.0f / fmaxf(deg[row], 1.0f);
        const float* xr = x       + (size_t)row * D_IN;
        const float* sr = support + (size_t)row * D_IN;
        int c0 = cg * 16;
        #pragma unroll
        for (int c = 0; c < 16; ++c) {
            int col = c0 + c;
            float v = 0.0f;
            if (valid) v = (col < D_IN) ? xr[col] : sr[col - D_IN] * invd;
            hs[r][col] = v;
        }
    }
    __syncthreads();

    // ---- Stage 2: 16x16 output tile per wave via V_WMMA_F32_16X16X4_F32 ----
    const int col0  = wave * 16;
    const int l16   = lane & 15;
    const int khalf = (lane >> 4) * 2;  // lanes 0-15 -> K+0/K+1, 16-31 -> K+2/K+3

    v8f acc = {};
    #pragma unroll 4
    for (int k = 0; k < D_HID; k += 4) {
        v2f a, b;
        // A (16x4, M x K): lane l16 holds row M=l16; VGPR0=K+khalf, VGPR1=K+khalf+1.
        a.x = hs[l16][k + khalf + 0];
        a.y = hs[l16][k + khalf + 1];
        // B (4x16, K x N): lane l16 holds col N=col0+l16, mirrored K split.
        b.x = W[(k + khalf + 0) * D_OUT + col0 + l16];
        b.y = W[(k + khalf + 1) * D_OUT + col0 + l16];
        acc = __builtin_amdgcn_wmma_f32_16x16x4_f32(
            /*neg_a=*/false, a, /*neg_b=*/false, b,
            /*c_mod=*/(short)0, acc, /*reuse_a=*/false, /*reuse_b=*/false);
    }

    // ---- Stage 3: bias + store per C/D layout (VGPR j -> rows j / j+8) ----
    float bv = bias[col0 + l16];
    int orow = row0 + (lane >> 4) * 8;
    if (row0 + 16 <= n) {
        #pragma unroll
        for (int j = 0; j < 8; ++j)
            out[(size_t)(orow + j) * D_OUT + col0 + l16] = acc[j] + bv;
    } else {
        #pragma unroll
        for (int j = 0; j < 8; ++j) {
            int rr = orow + j;
            if (rr < n) out[(size_t)rr * D_OUT + col0 + l16] = acc[j] + bv;
        }
    }
}

// ---------------------------------------------------------------------------
extern "C" void kernel_launch(void* const* d_in, const int* in_sizes, int n_in,
                              void* d_out, int out_size, void* d_ws, size_t ws_size,
                              hipStream_t stream) {
    const float* x    = (const float*)d_in[0];   // [N, 64]
    const float* W    = (const float*)d_in[1];   // [128, 64]
    const float* bias = (const float*)d_in[2];   // [1, 64]
    const int*   ei   = (const int*)d_in[3];     // [2, E] int32

    const int n_nodes = in_sizes[0] / D_IN;
    const int n_edges = in_sizes[3] / 2;

    float* support = (float*)d_ws;                       // [N, 64]
    float* deg     = support + (size_t)n_nodes * D_IN;   // [N]
    float* out     = (float*)d_out;

    // 1) zero support + deg (contiguous 65 floats per node)
    int zcount = n_nodes * (D_IN + 1);
    sage_zero<<<(zcount + 255) / 256, 256, 0, stream>>>(support, zcount);

    // 2) scatter-add: one edge per wave32 (8 edges per 256-thread block)
    long long sthreads = (long long)n_edges * 32;
    int sblocks = (int)((sthreads + 255) / 256);
    sage_scatter<<<sblocks, 256, 0, stream>>>(ei, x, support, deg, n_edges);

    // 3) fused mean/concat/GEMM, 16-row tile per 128-thread block
    int ntiles = (n_nodes + 15) / 16;
    sage_gemm<<<ntiles, 128, 0, stream>>>(x, W, bias, support, deg, out, n_nodes);
}